// CNNLayer_67482526155222
// MI455X (gfx1250) — compile-verified
//
#include <hip/hip_runtime.h>

typedef float v2f __attribute__((ext_vector_type(2)));
typedef float v8f __attribute__((ext_vector_type(8)));

// ---------------------------------------------------------------------------
// Stage 0: stride-8 "interpolation" conv: [256,8192,1] -> [256*1024, 16]
// y[row, ch] = relu(sum_{k<8} x[row*8+k] * f[k,ch] + bias[ch])
// Tiny (67 MFLOP): plain VALU kernel.
// ---------------------------------------------------------------------------
__global__ __launch_bounds__(256) void k_in_conv(
    const float* __restrict__ x, const float* __restrict__ f,
    const float* __restrict__ bias, float* __restrict__ y, int nrows)
{
  int idx = blockIdx.x * blockDim.x + threadIdx.x;
  if (idx >= nrows * 16) return;
  int ch  = idx & 15;
  int row = idx >> 4;
  const float* xp = x + (size_t)row * 8;
  float acc = bias[ch];
#pragma unroll
  for (int k = 0; k < 8; ++k) acc = fmaf(xp[k], f[k * 16 + ch], acc);
  y[idx] = fmaxf(acc, 0.0f);
}

// ---------------------------------------------------------------------------
// Main butterfly level as GEMM + bias + relu using V_WMMA_F32_16X16X4_F32.
//   X:[M,K] row-major, W:[K,N] row-major, Y:[M,N] row-major.
// One wave -> (IT*16) x (JT*16) output tile; K consumed 4 at a time.
// f32 WMMA lane layout (ISA 7.12.2):
//   A 16x4 : lane(l): M = l&15, holds K = {0,1} (l<16) or {2,3} (l>=16) in v2f
//   B 4x16 : lane(l): N = l&15, vgpr0 = row K=lk, vgpr1 = row K=lk+1
//   C 16x16: vgpr r : M = r + 8*(l>=16), N = l&15
// ---------------------------------------------------------------------------
template <int IT, int JT>
__global__ __launch_bounds__(256) void k_gemm_relu(
    const float* __restrict__ X, const float* __restrict__ W,
    const float* __restrict__ bias, float* __restrict__ Y,
    int M, int N, int K, int tiles_n)
{
  const int wid     = blockIdx.x * 8 + (threadIdx.x >> 5);
  const int tiles_m = M / (IT * 16);
  if (wid >= tiles_m * tiles_n) return;           // uniform per-wave: EXEC stays all-1s
  const int tm   = wid / tiles_n;
  const int tn   = wid - tm * tiles_n;
  const int m0   = tm * (IT * 16);
  const int n0   = tn * 64;
  const int lane = threadIdx.x & 31;
  const int lm   = lane & 15;
  const int lk   = (lane >> 4) << 1;              // 0 or 2

  v8f acc[IT][JT] = {};

  const float* Xp = X + (size_t)(m0 + lm) * K + lk;     // A: 8B-aligned v2f loads
  const float* Wp = W + (size_t)lk * N + n0 + lm;

  for (int kk = 0; kk < K; kk += 4) {
    v2f a[IT];
#pragma unroll
    for (int i = 0; i < IT; ++i)
      a[i] = *(const v2f*)(Xp + (size_t)i * 16 * K + kk);

    v2f b[JT];
#pragma unroll
    for (int j = 0; j < JT; ++j) {
      const float* wp = Wp + (size_t)kk * N + j * 16;
      b[j].x = wp[0];          // row kk+lk
      b[j].y = wp[N];          // row kk+lk+1
    }

#pragma unroll
    for (int i = 0; i < IT; ++i)
#pragma unroll
      for (int j = 0; j < JT; ++j)
        acc[i][j] = __builtin_amdgcn_wmma_f32_16x16x4_f32(
            false, a[i], false, b[j], (short)0, acc[i][j], false, false);
  }

  const int rsel = (lane >> 4) << 3;   // lanes >=16 hold rows +8
#pragma unroll
  for (int j = 0; j < JT; ++j) {
    const int   col = n0 + j * 16 + lm;
    const float bv  = bias[col];
#pragma unroll
    for (int i = 0; i < IT; ++i) {
#pragma unroll
      for (int r = 0; r < 8; ++r) {
        const int row = m0 + i * 16 + rsel + r;
        Y[(size_t)row * N + col] = fmaxf(acc[i][j][r] + bv, 0.0f);
      }
    }
  }
}

// ---------------------------------------------------------------------------
// Final per-block matmul: feat [256(batch), 256(blocks)*16] viewed [b,k,c];
// out[b,k,o] = sum_c feat[b,k,c] * fd[k,c,o].  One wave per (k, m-tile):
// [16 batch rows, 16] x [16,16] via 4 f32 WMMAs.  4096 waves = 512 blocks.
// ---------------------------------------------------------------------------
__global__ __launch_bounds__(256) void k_fea(
    const float* __restrict__ feat, const float* __restrict__ fd,
    float* __restrict__ out)
{
  const int wid  = blockIdx.x * 8 + (threadIdx.x >> 5);  // 0..4095
  const int kblk = wid >> 4;                              // block index 0..255
  const int mt   = wid & 15;                              // batch tile 0..15
  const int lane = threadIdx.x & 31;
  const int lm   = lane & 15;
  const int lk   = (lane >> 4) << 1;

  const float* A = feat + (size_t)(mt * 16 + lm) * 4096 + kblk * 16 + lk;
  const float* B = fd + (size_t)kblk * 256 + lm;

  v8f c = {};
#pragma unroll
  for (int kk = 0; kk < 16; kk += 4) {
    v2f a = *(const v2f*)(A + kk);
    v2f b;
    b.x = B[(kk + lk) * 16];
    b.y = B[(kk + lk + 1) * 16];
    c = __builtin_amdgcn_wmma_f32_16x16x4_f32(
        false, a, false, b, (short)0, c, false, false);
  }

  const int rsel = (lane >> 4) << 3;
  const int col  = kblk * 16 + lm;
#pragma unroll
  for (int r = 0; r < 8; ++r)
    out[(size_t)(mt * 16 + rsel + r) * 4096 + col] = c[r];   // no relu on output
}

// ---------------------------------------------------------------------------
// Launch: in_conv -> 10 WMMA GEMM levels (ping-pong in d_ws) -> block matmul.
// d_in order: in_data, in_filter, in_bias, filters[0..9], biases[0..9], fea_dense
// ---------------------------------------------------------------------------
extern "C" void kernel_launch(void* const* d_in, const int* in_sizes, int n_in,
                              void* d_out, int out_size, void* d_ws, size_t ws_size,
                              hipStream_t stream) {
  (void)in_sizes; (void)n_in; (void)out_size; (void)ws_size;

  const float* in_data   = (const float*)d_in[0];
  const float* in_filter = (const float*)d_in[1];
  const float* in_bias   = (const float*)d_in[2];
  const float* filt[10];
  const float* bs[10];
  for (int l = 0; l < 10; ++l) {
    filt[l] = (const float*)d_in[3 + l];
    bs[l]   = (const float*)d_in[13 + l];
  }
  const float* fea_dense = (const float*)d_in[23];
  float* out = (float*)d_out;

  float* buf0 = (float*)d_ws;                 // 16 MB half
  float* buf1 = buf0 + (size_t)(4u << 20);    // second 16 MB half

  {
    const int nrows = 256 * 1024;
    const int threads = nrows * 16;
    k_in_conv<<<(threads + 255) / 256, 256, 0, stream>>>(
        in_data, in_filter, in_bias, buf0, nrows);
  }

  const float* src = buf0;
  float* dst = buf1;
  for (int l = 0; l < 10; ++l) {               // level = l+1
    const int Lout = 1024 >> (l + 1);
    const int cin  = 16 << (l < 8 ? l : 8);
    const int cout = 16 << ((l + 1) < 8 ? (l + 1) : 8);
    const int M = 256 * Lout;
    const int K = 2 * cin;
    const int N = cout;
    const int tiles_n = (N + 63) >> 6;
    const int IT = (M >= 1024) ? 4 : ((M >= 512) ? 2 : 1);
    const int tiles_m = M / (IT * 16);
    const int blocks = (tiles_m * tiles_n + 7) / 8;

    if (N >= 64) {
      if (IT == 4)
        k_gemm_relu<4,4><<<blocks,256,0,stream>>>(src, filt[l], bs[l], dst, M, N, K, tiles_n);
      else if (IT == 2)
        k_gemm_relu<2,4><<<blocks,256,0,stream>>>(src, filt[l], bs[l], dst, M, N, K, tiles_n);
      else
        k_gemm_relu<1,4><<<blocks,256,0,stream>>>(src, filt[l], bs[l], dst, M, N, K, tiles_n);
    } else {   // level 1: N = 32
      k_gemm_relu<4,2><<<blocks,256,0,stream>>>(src, filt[l], bs[l], dst, M, N, K, tiles_n);
    }
    const float* t = src; src = dst; dst = (float*)t;
  }

  // src now holds [256, 4096] features (buf0 after 10 swaps)
  k_fea<<<512, 256, 0, stream>>>(src, fea_dense, out);
}